// ComplexLoss_14620068676244
// MI455X (gfx1250) — compile-verified
//
#include <hip/hip_runtime.h>

// ComplexLoss on MI455X (gfx1250).
// Bandwidth-bound streaming reduction: ~160 MB traffic -> ~7us roofline at 23.3 TB/s.
// No matmul structure => WMMA not applicable; we exercise the CDNA5 data-movement
// paths instead: async global->LDS staging (ASYNCcnt + s_wait_asynccnt),
// global_prefetch_b8, wave32 ballot/head-flag segmented reduction, few global atomics.

#define K_SEG   8192
#define BLOCK   256
#define ITER    8
#define TILE    (BLOCK * ITER)   // 2048 nodes per block tile
#define ALPHA_W 0.5f

// ---------------------------------------------------------------- init -----
__global__ __launch_bounds__(BLOCK)
void cl_init_ws(float* __restrict__ segSum,
                unsigned* __restrict__ segMax,
                unsigned* __restrict__ segCnt) {
    int i = blockIdx.x * BLOCK + threadIdx.x;
    if (i < K_SEG) {
        segSum[i] = 0.0f;
        segMax[i] = 0u;      // bits of +0.0f; node_loss >= 0 so this is a valid floor
        segCnt[i] = 0u;
    }
}

// ---------------------------------------------------------------- main -----
__global__ __launch_bounds__(BLOCK)
void cl_focal_seg(const float* __restrict__ logits,
                  const int*   __restrict__ targets,
                  const int*   __restrict__ cid,
                  float*       __restrict__ segSum,
                  unsigned*    __restrict__ segMax,
                  unsigned*    __restrict__ segCnt,
                  int N) {
    __shared__ int ldsId[TILE];

    const int tid  = threadIdx.x;
    const int lane = tid & 31;
    const long long tileStart = (long long)blockIdx.x * TILE;

    // ---- Stage this tile's complex ids into LDS with gfx1250 async copies.
    // Thread t stages entries {i*BLOCK + t}; at iteration i, wave w reads entries
    // i*BLOCK + [32w,32w+32) which were staged by the *same* wave's lanes, so a
    // per-wave s_wait_asynccnt(0) is sufficient -- no workgroup barrier needed.
#pragma unroll
    for (int i = 0; i < ITER; ++i) {
        int idx = i * BLOCK + tid;
        long long node = tileStart + idx;
        if (node >= N) node = N - 1;              // clamp: safe read, ignored later
        const int* gp = cid + node;
        // Low 32 bits of a generic pointer to LDS == wave-relative LDS byte offset.
        unsigned ldsAddr = (unsigned)(unsigned long long)(&ldsId[idx]);
        asm volatile("global_load_async_to_lds_b32 %0, %1, off"
                     :: "v"(ldsAddr), "v"(gp) : "memory");
    }
    asm volatile("s_wait_asynccnt 0" ::: "memory");

#pragma unroll
    for (int i = 0; i < ITER; ++i) {
        long long node = tileStart + i * BLOCK + tid;
        bool valid = (node < (long long)N);
        long long nn = valid ? node : (long long)(N - 1);

        // Prefetch logits a few iterations ahead (emits global_prefetch_b8).
        long long pfn = tileStart + (i + 4) * BLOCK + tid;
        if (pfn < (long long)N)
            __builtin_prefetch(logits + pfn * 8, 0, 3);

        // 32B/node via two coalesced b128 loads.
        const float4* lp = (const float4*)(logits + nn * 8);
        float4 a = lp[0];
        float4 b = lp[1];
        int tgt = targets[nn];

        // log-sum-exp over C=8 (fast-math exp/log: bare v_exp_f32 / v_log_f32;
        // ~1e-6 rel error per node, negligible after 4M-node averaging)
        float m = fmaxf(fmaxf(fmaxf(a.x, a.y), fmaxf(a.z, a.w)),
                        fmaxf(fmaxf(b.x, b.y), fmaxf(b.z, b.w)));
        float s = __expf(a.x - m) + __expf(a.y - m) + __expf(a.z - m) + __expf(a.w - m)
                + __expf(b.x - m) + __expf(b.y - m) + __expf(b.z - m) + __expf(b.w - m);
        float xt = (tgt == 0) ? a.x : (tgt == 1) ? a.y : (tgt == 2) ? a.z :
                   (tgt == 3) ? a.w : (tgt == 4) ? b.x : (tgt == 5) ? b.y :
                   (tgt == 6) ? b.z : b.w;
        // s >= 1 (contains exp(0) term) => log(s) >= 0; m - xt >= 0  => ce >= 0.
        float ce = (m - xt) + __logf(s);
        float pt = __expf(-ce);
        float om = 1.0f - pt;                 // >= 0
        float loss = om * om * ce;            // GAMMA = 2; loss >= 0

        int id = ldsId[i * BLOCK + tid];
        if (!valid) { id = -1; loss = 0.0f; }

        // ---- wave32 segmented reduction over sorted ids (head-flag form) --
        // One id shuffle + ballot gives head positions; the scan guard is
        // purely positional, and run length falls out for free.
        int prevId = __shfl_up(id, 1, 32);
        bool head = (lane == 0) || (prevId != id);
        unsigned heads = (unsigned)__ballot(head);          // wave32: low 32 bits
        unsigned mask_le = 0xFFFFFFFFu >> (31 - lane);      // bits [0..lane]
        int headpos = 31 - __clz((int)(heads & mask_le));   // my run's first lane
        unsigned tails = (heads >> 1) | 0x80000000u;        // tail = next lane is head
        bool tail = (tails >> lane) & 1u;
        int cnt = lane - headpos + 1;                       // my run length so far

        float vS = loss;
        float vM = loss;
#pragma unroll
        for (int off = 1; off < 32; off <<= 1) {
            float oS = __shfl_up(vS, off, 32);
            float oM = __shfl_up(vM, off, 32);
            if (lane - off >= headpos) {        // source lane inside my run
                vS += oS;
                vM = fmaxf(vM, oM);
            }
        }

        if (tail && id >= 0) {
            atomicAdd(&segSum[id], vS);
            // bit pattern of non-negative f32 is monotone as u32
            atomicMax(&segMax[id], __float_as_uint(vM));
            atomicAdd(&segCnt[id], (unsigned)cnt);
        }
    }
}

// ------------------------------------------------------------- finalize ----
__global__ __launch_bounds__(BLOCK)
void cl_finalize(const float* __restrict__ segSum,
                 const unsigned* __restrict__ segMax,
                 const unsigned* __restrict__ segCnt,
                 float* __restrict__ out) {
    __shared__ float rs[BLOCK];
    __shared__ float rn[BLOCK];
    const int tid = threadIdx.x;

    float acc = 0.0f, np = 0.0f;
    for (int k = tid; k < K_SEG; k += BLOCK) {
        unsigned c = segCnt[k];
        if (c > 0u) {
            float mx = __uint_as_float(segMax[k]);
            acc += ALPHA_W * segSum[k] + (1.0f - ALPHA_W) * mx;
            np  += 1.0f;
        }
    }
    rs[tid] = acc;
    rn[tid] = np;
    __syncthreads();
#pragma unroll
    for (int off = BLOCK / 2; off > 0; off >>= 1) {
        if (tid < off) {
            rs[tid] += rs[tid + off];
            rn[tid] += rn[tid + off];
        }
        __syncthreads();
    }
    if (tid == 0)
        out[0] = rs[0] / fmaxf(rn[0], 1.0f);
}

// --------------------------------------------------------------- launch ----
extern "C" void kernel_launch(void* const* d_in, const int* in_sizes, int n_in,
                              void* d_out, int out_size, void* d_ws, size_t ws_size,
                              hipStream_t stream) {
    const float* logits  = (const float*)d_in[0];
    const int*   targets = (const int*)d_in[1];
    const int*   cid     = (const int*)d_in[2];
    const int N = in_sizes[1];   // targets count == N

    float*    segSum = (float*)d_ws;
    unsigned* segMax = (unsigned*)((float*)d_ws + K_SEG);
    unsigned* segCnt = segMax + K_SEG;

    cl_init_ws<<<(K_SEG + BLOCK - 1) / BLOCK, BLOCK, 0, stream>>>(segSum, segMax, segCnt);

    int nBlocks = (N + TILE - 1) / TILE;
    cl_focal_seg<<<nBlocks, BLOCK, 0, stream>>>(logits, targets, cid,
                                                segSum, segMax, segCnt, N);

    cl_finalize<<<1, BLOCK, 0, stream>>>(segSum, segMax, segCnt, (float*)d_out);
}